// MultiLevelEncoder_37744172597585
// MI455X (gfx1250) — compile-verified
//
#include <hip/hip_runtime.h>

#define D_MODEL 512
#define D_INNER 1024
#define D_STATE 16
#define D_CONV  4
#define DT_RANK 32
#define N_LAYERS 3
#define BATCH 4
#define SEQ 1024
#define BS (BATCH * SEQ)                 // 4096 activation rows
#define D_PROJ (DT_RANK + 2 * D_STATE)   // 64
#define SCAN_PF 8                        // scan prefetch distance (timesteps)

typedef __attribute__((ext_vector_type(2))) float v2f;
typedef __attribute__((ext_vector_type(8))) float v8f;

__device__ __forceinline__ float sigmoid_f(float x) { return 1.f / (1.f + __expf(-x)); }
__device__ __forceinline__ float softplus_f(float x) { return x > 20.f ? x : log1pf(__expf(x)); }

// -------------------------------------------------------------------------
// f32 WMMA GEMM: Y[M,N] = act( A[M,K](lda) * W[N,K](ldw)^T + bias[N] ) + addend
// One wave (32 threads) per block computes a 32x64 tile (2 M-tiles x 4 N-tiles)
// via V_WMMA_F32_16X16X4_F32, stepping K by 4. grid = (N/64, M/32).
// ACT: 0 = none, 1 = softplus.
// -------------------------------------------------------------------------
template <int ACT>
__global__ __launch_bounds__(32) void gemm_wmma_f32(
    const float* __restrict__ A, int lda,
    const float* __restrict__ W, int ldw,
    const float* __restrict__ bias,
    const float* __restrict__ addend,
    float* __restrict__ Y, int ldy, int K)
{
  const int lane = threadIdx.x & 31;
  const int half = lane >> 4;   // 0: K lanes {0,1}, 1: K lanes {2,3}
  const int r    = lane & 15;   // M (for A) / N (for B) index within tile
  const int i0 = blockIdx.y * 32;
  const int j0 = blockIdx.x * 64;

  v8f acc[2][4];
#pragma unroll
  for (int mt = 0; mt < 2; ++mt)
#pragma unroll
    for (int nt = 0; nt < 4; ++nt)
#pragma unroll
      for (int v = 0; v < 8; ++v) acc[mt][nt][v] = 0.f;

  const float* __restrict__ Ar[2] = {A + (size_t)(i0 + r) * lda,
                                     A + (size_t)(i0 + 16 + r) * lda};
  const float* __restrict__ Wr[4];
#pragma unroll
  for (int nt = 0; nt < 4; ++nt) Wr[nt] = W + (size_t)(j0 + nt * 16 + r) * ldw;

  for (int k = 0; k < K; k += 4) {
    const int kk = k + 2 * half;
    v2f a[2], b[4];
#pragma unroll
    for (int mt = 0; mt < 2; ++mt) a[mt] = *(const v2f*)(Ar[mt] + kk);
#pragma unroll
    for (int nt = 0; nt < 4; ++nt) b[nt] = *(const v2f*)(Wr[nt] + kk);
#pragma unroll
    for (int mt = 0; mt < 2; ++mt)
#pragma unroll
      for (int nt = 0; nt < 4; ++nt)
        acc[mt][nt] = __builtin_amdgcn_wmma_f32_16x16x4_f32(
            false, a[mt], false, b[nt], (short)0, acc[mt][nt], false, false);
  }

  // C/D layout: VGPR v <-> M = v + 8*half, N = r (per 16x16 tile)
#pragma unroll
  for (int mt = 0; mt < 2; ++mt) {
#pragma unroll
    for (int nt = 0; nt < 4; ++nt) {
      const int n = j0 + nt * 16 + r;
      const float bv = bias ? bias[n] : 0.f;
#pragma unroll
      for (int v = 0; v < 8; ++v) {
        const int m = i0 + mt * 16 + half * 8 + v;
        float val = acc[mt][nt][v] + bv;
        if (ACT == 1) val = softplus_f(val);
        if (addend) val += addend[(size_t)m * ldy + n];
        Y[(size_t)m * ldy + n] = val;
      }
    }
  }
}

// ---------------- rmsnorm: one wave per row (D_MODEL=512), float4 ---------
__global__ void rmsnorm_kernel(const float* __restrict__ x,
                               const float* __restrict__ w,
                               float* __restrict__ y, int rows)
{
  const int lane = threadIdx.x & 31;
  const int row  = blockIdx.x * 8 + (threadIdx.x >> 5);
  if (row >= rows) return;
  const float4* xr = (const float4*)(x + (size_t)row * D_MODEL);
  const float4* wr = (const float4*)w;
  float4* yr = (float4*)(y + (size_t)row * D_MODEL);

  float4 v[4];
  float ss = 0.f;
#pragma unroll
  for (int q = 0; q < 4; ++q) {
    v[q] = xr[lane + 32 * q];
    ss += v[q].x * v[q].x + v[q].y * v[q].y + v[q].z * v[q].z + v[q].w * v[q].w;
  }
#pragma unroll
  for (int off = 16; off; off >>= 1) ss += __shfl_xor(ss, off, 32);
  const float scale = rsqrtf(ss * (1.f / D_MODEL) + 1e-5f);
#pragma unroll
  for (int q = 0; q < 4; ++q) {
    const float4 wv = wr[lane + 32 * q];
    float4 o;
    o.x = v[q].x * scale * wv.x;
    o.y = v[q].y * scale * wv.y;
    o.z = v[q].z * scale * wv.z;
    o.w = v[q].w * scale * wv.w;
    yr[lane + 32 * q] = o;
  }
}

// -------- depthwise causal conv (k=4) + SiLU; xi = low half of xz ---------
__global__ void conv_silu_kernel(const float* __restrict__ xz,
                                 const float* __restrict__ cw,
                                 const float* __restrict__ cb,
                                 float* __restrict__ xc)
{
  const int gid = blockIdx.x * blockDim.x + threadIdx.x; // BS * D_INNER
  const int d  = gid & (D_INNER - 1);
  const int bs = gid >> 10;
  const int s  = bs & (SEQ - 1);
  float acc = cb[d];
#pragma unroll
  for (int j = 0; j < D_CONV; ++j) {
    const int sp = s - (D_CONV - 1) + j;
    if (sp >= 0)
      acc += cw[d * D_CONV + j] *
             xz[(size_t)(bs - (D_CONV - 1) + j) * (2 * D_INNER) + d];
  }
  xc[gid] = acc * sigmoid_f(acc);
}

// -------------------------------------------------------------------------
// Selective scan: thread per (b,d); 16-state in registers; y written over u.
// Software-pipelined: iteration t+1's loads issue before iteration t's store
// (same-array store would otherwise block load hoisting). global_prefetch_b8
// warms WGP$ SCAN_PF timesteps ahead for the latency-bound serial loop.
// -------------------------------------------------------------------------
__global__ void scan_kernel(float* __restrict__ u_y,           // in: u, out: y
                            const float* __restrict__ delta,   // [B,S,D_INNER]
                            const float* __restrict__ proj,    // [B,S,64]
                            const float* __restrict__ A_log,   // [D_INNER,16]
                            const float* __restrict__ Dskip)   // [D_INNER]
{
  const int gid = blockIdx.x * blockDim.x + threadIdx.x; // BATCH * D_INNER
  const int b = gid >> 10;
  const int d = gid & (D_INNER - 1);
  float Ac[D_STATE];
#pragma unroll
  for (int n = 0; n < D_STATE; ++n) Ac[n] = -__expf(A_log[d * D_STATE + n]);
  const float Dv = Dskip[d];
  float h[D_STATE];
#pragma unroll
  for (int n = 0; n < D_STATE; ++n) h[n] = 0.f;

  size_t idx  = (size_t)(b * SEQ) * D_INNER + d;
  size_t prow = (size_t)(b * SEQ) * D_PROJ;

  // ---- prologue: load t = 0 ----
  float dl = delta[idx];
  float u  = u_y[idx];
  float4 Bv[4], Cv[4];
#pragma unroll
  for (int q = 0; q < 4; ++q) {
    Bv[q] = ((const float4*)(proj + prow + DT_RANK))[q];
    Cv[q] = ((const float4*)(proj + prow + DT_RANK + D_STATE))[q];
  }

  for (int t = 0; t < SEQ - 1; ++t) {
    // prefetch SCAN_PF steps ahead (no counter cost; pulls lines toward WGP$)
    if (t + SCAN_PF < SEQ) {
      __builtin_prefetch(delta + idx + (size_t)SCAN_PF * D_INNER, 0, 1);
      __builtin_prefetch(u_y   + idx + (size_t)SCAN_PF * D_INNER, 0, 1);
      __builtin_prefetch(proj  + prow + (size_t)SCAN_PF * D_PROJ, 0, 1);
    }
    // ---- issue t+1 loads before t's store ----
    const size_t idx_n  = idx + D_INNER;
    const size_t prow_n = prow + D_PROJ;
    const float dl_n = delta[idx_n];
    const float u_n  = u_y[idx_n];
    float4 Bn[4], Cn[4];
#pragma unroll
    for (int q = 0; q < 4; ++q) {
      Bn[q] = ((const float4*)(proj + prow_n + DT_RANK))[q];
      Cn[q] = ((const float4*)(proj + prow_n + DT_RANK + D_STATE))[q];
    }
    // ---- compute t ----
    const float du = dl * u;
    float y = 0.f;
#pragma unroll
    for (int q = 0; q < 4; ++q) {
      const int n0 = q * 4;
      h[n0+0] = __expf(dl * Ac[n0+0]) * h[n0+0] + du * Bv[q].x;  y += h[n0+0] * Cv[q].x;
      h[n0+1] = __expf(dl * Ac[n0+1]) * h[n0+1] + du * Bv[q].y;  y += h[n0+1] * Cv[q].y;
      h[n0+2] = __expf(dl * Ac[n0+2]) * h[n0+2] + du * Bv[q].z;  y += h[n0+2] * Cv[q].z;
      h[n0+3] = __expf(dl * Ac[n0+3]) * h[n0+3] + du * Bv[q].w;  y += h[n0+3] * Cv[q].w;
    }
    u_y[idx] = y + u * Dv;
    // ---- rotate ----
    dl = dl_n; u = u_n;
#pragma unroll
    for (int q = 0; q < 4; ++q) { Bv[q] = Bn[q]; Cv[q] = Cn[q]; }
    idx = idx_n; prow = prow_n;
  }
  // ---- epilogue: t = SEQ-1 ----
  const float du = dl * u;
  float y = 0.f;
#pragma unroll
  for (int q = 0; q < 4; ++q) {
    const int n0 = q * 4;
    h[n0+0] = __expf(dl * Ac[n0+0]) * h[n0+0] + du * Bv[q].x;  y += h[n0+0] * Cv[q].x;
    h[n0+1] = __expf(dl * Ac[n0+1]) * h[n0+1] + du * Bv[q].y;  y += h[n0+1] * Cv[q].y;
    h[n0+2] = __expf(dl * Ac[n0+2]) * h[n0+2] + du * Bv[q].z;  y += h[n0+2] * Cv[q].z;
    h[n0+3] = __expf(dl * Ac[n0+3]) * h[n0+3] + du * Bv[q].w;  y += h[n0+3] * Cv[q].w;
  }
  u_y[idx] = y + u * Dv;
}

// ---------------- gating: g = y * silu(z), z = high half of xz ------------
__global__ void gate_kernel(const float* __restrict__ y,
                            const float* __restrict__ xz,
                            float* __restrict__ g)
{
  const int gid = blockIdx.x * blockDim.x + threadIdx.x;
  const int d  = gid & (D_INNER - 1);
  const int bs = gid >> 10;
  const float z = xz[(size_t)bs * (2 * D_INNER) + D_INNER + d];
  g[gid] = y[gid] * z * sigmoid_f(z);
}

// ---------------- attention mask: (sum(input,-1) == padding_idx) ----------
__global__ void mask_kernel(const float* __restrict__ in,
                            const int* __restrict__ pad,
                            float* __restrict__ out, int rows)
{
  const int lane = threadIdx.x & 31;
  const int row  = blockIdx.x * 8 + (threadIdx.x >> 5);
  if (row >= rows) return;
  const float4* xr = (const float4*)(in + (size_t)row * D_MODEL);
  float s = 0.f;
#pragma unroll
  for (int q = 0; q < 4; ++q) {
    const float4 v = xr[lane + 32 * q];
    s += v.x + v.y + v.z + v.w;
  }
#pragma unroll
  for (int off = 16; off; off >>= 1) s += __shfl_xor(s, off, 32);
  if (lane == 0) out[row] = (s == (float)(*pad)) ? 1.f : 0.f;
}

extern "C" void kernel_launch(void* const* d_in, const int* in_sizes, int n_in,
                              void* d_out, int out_size, void* d_ws, size_t ws_size,
                              hipStream_t stream)
{
  (void)in_sizes; (void)n_in; (void)out_size; (void)ws_size;
  const float* input = (const float*)d_in[0];
  const float* ipw   = (const float*)d_in[1];
  const float* cw    = (const float*)d_in[2];
  const float* cb    = (const float*)d_in[3];
  const float* xpw   = (const float*)d_in[4];
  const float* dpw   = (const float*)d_in[5];
  const float* dpb   = (const float*)d_in[6];
  const float* alog  = (const float*)d_in[7];
  const float* dsk   = (const float*)d_in[8];
  const float* opw   = (const float*)d_in[9];
  const float* nw    = (const float*)d_in[10];
  const float* nfw   = (const float*)d_in[11];
  const int*   pad   = (const int*)d_in[12];

  // Workspace layout (floats): ~85 MB total, L2-resident on MI455X (192 MB L2)
  float* ws   = (float*)d_ws;
  float* xcur = ws;                                  // BS*D_MODEL   residual stream
  float* xn   = xcur + (size_t)BS * D_MODEL;         // BS*D_MODEL   normed x
  float* xz   = xn   + (size_t)BS * D_MODEL;         // BS*2*D_INNER in_proj out
  float* xc   = xz   + (size_t)BS * 2 * D_INNER;     // BS*D_INNER   conv out -> scan y
  float* proj = xc   + (size_t)BS * D_INNER;         // BS*64        dt|B|C
  float* dlt  = proj + (size_t)BS * D_PROJ;          // BS*D_INNER   delta -> gated g

  hipMemcpyAsync(xcur, input, (size_t)BS * D_MODEL * sizeof(float),
                 hipMemcpyDeviceToDevice, stream);

  const dim3 blk32(32);
  for (int l = 0; l < N_LAYERS; ++l) {
    rmsnorm_kernel<<<BS / 8, 256, 0, stream>>>(xcur, nw + (size_t)l * D_MODEL, xn, BS);

    // xz = xn @ in_proj_w^T   (M=4096, N=2048, K=512)
    gemm_wmma_f32<0><<<dim3(2 * D_INNER / 64, BS / 32), blk32, 0, stream>>>(
        xn, D_MODEL, ipw + (size_t)l * 2 * D_INNER * D_MODEL, D_MODEL,
        nullptr, nullptr, xz, 2 * D_INNER, D_MODEL);

    conv_silu_kernel<<<(BS * D_INNER) / 256, 256, 0, stream>>>(
        xz, cw + (size_t)l * D_INNER * D_CONV, cb + (size_t)l * D_INNER, xc);

    // proj = xc @ x_proj_w^T  (M=4096, N=64, K=1024)
    gemm_wmma_f32<0><<<dim3(D_PROJ / 64, BS / 32), blk32, 0, stream>>>(
        xc, D_INNER, xpw + (size_t)l * D_PROJ * D_INNER, D_INNER,
        nullptr, nullptr, proj, D_PROJ, D_INNER);

    // delta = softplus(dt @ dt_proj_w^T + dt_proj_b)  (M=4096, N=1024, K=32)
    gemm_wmma_f32<1><<<dim3(D_INNER / 64, BS / 32), blk32, 0, stream>>>(
        proj, D_PROJ, dpw + (size_t)l * D_INNER * DT_RANK, DT_RANK,
        dpb + (size_t)l * D_INNER, nullptr, dlt, D_INNER, DT_RANK);

    scan_kernel<<<(BATCH * D_INNER) / 256, 256, 0, stream>>>(
        xc, dlt, proj, alog + (size_t)l * D_INNER * D_STATE,
        dsk + (size_t)l * D_INNER);

    gate_kernel<<<(BS * D_INNER) / 256, 256, 0, stream>>>(xc, xz, dlt);

    // xcur = g @ out_proj_w^T + xcur  (M=4096, N=512, K=1024), residual fused
    gemm_wmma_f32<0><<<dim3(D_MODEL / 64, BS / 32), blk32, 0, stream>>>(
        dlt, D_INNER, opw + (size_t)l * D_MODEL * D_INNER, D_INNER,
        nullptr, xcur, xcur, D_MODEL, D_INNER);
  }

  float* out = (float*)d_out;
  rmsnorm_kernel<<<BS / 8, 256, 0, stream>>>(xcur, nfw, out, BS);
  mask_kernel<<<BS / 8, 256, 0, stream>>>(input, pad, out + (size_t)BS * D_MODEL, BS);
}